// CostumeLoss_69947837383475
// MI455X (gfx1250) — compile-verified
//
#include <hip/hip_runtime.h>
#include <math.h>

// ---------------------------------------------------------------------------
// Contrastive / discriminative loss, single HBM pass.
//   Pass 1 (WMMA): counts[K], sums[K][D] = OneHot @ F, sumsq[K] = sum_d segsum(f^2)
//   Finalize:      var = sumsq/c - ||mu||^2 ; pairwise hinge ; reg ; sum/9
// ---------------------------------------------------------------------------

typedef __attribute__((ext_vector_type(2))) float v2f;
typedef __attribute__((ext_vector_type(8))) float v8f;

#define K_IDS 32
#define DIMS  32
#define NPIX  (512 * 512)
#define NIMG  8
#define DD_C    2.5f
#define GAMMA_C 0.005f

// workspace layout per image: [0..31]=counts [32..63]=sumsq [64..1087]=sums[k][d]
#define WS_PER_IMG 1088

#define BLOCKS_PER_IMG 64
#define CHUNK    (NPIX / BLOCKS_PER_IMG)   // 4096 pixels per block
#define WAVES    8                         // 256 threads, wave32
#define WAVE_PIX (CHUNK / WAVES)           // 512 pixels per wave
#define KSTEPS   (WAVE_PIX / 4)            // 128 k-steps (4 pixels each)

__global__ __launch_bounds__(256) void zero_ws_kernel(float* ws, int n) {
    int i = blockIdx.x * blockDim.x + threadIdx.x;
    if (i < n) ws[i] = 0.0f;
}

// ---------------------------------------------------------------------------
// Pass 1: one-hot WMMA segment sum + LDS-accumulated counts / sum-of-squares.
// Each wave: 512 contiguous pixels, 4 pixels per k-step.
//   C[16 inst,16 dim] += A[16 inst,4 pix] x B[4 pix,16 dim]   (f32 16x16x4)
// A/B K-slot mapping (ISA 7.12.2, 32-bit operands): k = vgpr_j + 2*(lane>=16).
// ---------------------------------------------------------------------------
__global__ __launch_bounds__(256) void segsum_wmma_kernel(
        const float* __restrict__ F,   // [NIMG][DIMS][NPIX]
        const int*   __restrict__ Lab, // [NIMG][NPIX]
        float*       __restrict__ ws)  // [NIMG][WS_PER_IMG]
{
    __shared__ float s_ssq[K_IDS * 16];   // [label][m] padded slots
    __shared__ float s_cnt[K_IDS * 16];

    const int tid = threadIdx.x;
    for (int i = tid; i < K_IDS * 16; i += blockDim.x) {
        s_ssq[i] = 0.0f;
        s_cnt[i] = 0.0f;
    }
    __syncthreads();

    const int img  = blockIdx.y;
    const int lane = tid & 31;
    const int wave = tid >> 5;
    const int half = lane >> 4;   // which 16-lane half
    const int m    = lane & 15;   // dim within tile / row within A tile

    const float* Fimg  = F   + (size_t)img * DIMS * NPIX;
    const int*   Limg  = Lab + (size_t)img * NPIX;
    const float* rowLo = Fimg + (size_t)m        * NPIX;  // dim m
    const float* rowHi = Fimg + (size_t)(m + 16) * NPIX;  // dim m+16

    const int wbase = blockIdx.x * CHUNK + wave * WAVE_PIX;

    v8f c00 = {}; v8f c01 = {}; v8f c10 = {}; v8f c11 = {};

#pragma unroll 2
    for (int t = 0; t < KSTEPS; ++t) {
        const int p = wbase + 4 * t;
        const int q = p + 2 * half;  // this lane's pixel pair (j=0 -> q, j=1 -> q+1)

        v2f  b0 = *(const v2f*)(rowLo + q);   // F[dim m   ][q], [q+1]
        v2f  b1 = *(const v2f*)(rowHi + q);   // F[dim m+16][q], [q+1]
        int2 lb = *(const int2*)(Limg + q);   // labels (broadcast within half)

        v2f a0, a1;                            // one-hot A tiles (exact in f32)
        a0.x = (lb.x == m)      ? 1.0f : 0.0f;
        a0.y = (lb.y == m)      ? 1.0f : 0.0f;
        a1.x = (lb.x == m + 16) ? 1.0f : 0.0f;
        a1.y = (lb.y == m + 16) ? 1.0f : 0.0f;

        c00 = __builtin_amdgcn_wmma_f32_16x16x4_f32(false, a0, false, b0,
                                                    (short)0, c00, false, false);
        c01 = __builtin_amdgcn_wmma_f32_16x16x4_f32(false, a0, false, b1,
                                                    (short)0, c01, false, false);
        c10 = __builtin_amdgcn_wmma_f32_16x16x4_f32(false, a1, false, b0,
                                                    (short)0, c10, false, false);
        c11 = __builtin_amdgcn_wmma_f32_16x16x4_f32(false, a1, false, b1,
                                                    (short)0, c11, false, false);

        // per-lane partial sum-of-squares for its two dims of each pixel
        float s0 = b0.x * b0.x + b1.x * b1.x;   // pixel q
        float s1 = b0.y * b0.y + b1.y * b1.y;   // pixel q+1
        atomicAdd(&s_ssq[lb.x * 16 + m], s0);
        atomicAdd(&s_cnt[lb.x * 16 + m], 1.0f); // 16 lanes/pixel -> /16 at flush
        atomicAdd(&s_ssq[lb.y * 16 + m], s1);
        atomicAdd(&s_cnt[lb.y * 16 + m], 1.0f);
    }

    // flush WMMA accumulators: C vgpr r holds rows r (lanes 0-15) / r+8 (16-31)
    float* sums = ws + (size_t)img * WS_PER_IMG + 64;
#pragma unroll
    for (int r = 0; r < 8; ++r) {
        const int i0 = r + 8 * half;  // local row within 16-row tile
        atomicAdd(&sums[(i0     ) * DIMS + (m     )], c00[r]);
        atomicAdd(&sums[(i0     ) * DIMS + (m + 16)], c01[r]);
        atomicAdd(&sums[(i0 + 16) * DIMS + (m     )], c10[r]);
        atomicAdd(&sums[(i0 + 16) * DIMS + (m + 16)], c11[r]);
    }

    __syncthreads();
    if (tid < K_IDS) {
        float s = 0.0f, c = 0.0f;
#pragma unroll
        for (int i = 0; i < 16; ++i) {
            s += s_ssq[tid * 16 + i];
            c += s_cnt[tid * 16 + i];
        }
        atomicAdd(&ws[(size_t)img * WS_PER_IMG + tid],      c * (1.0f / 16.0f));
        atomicAdd(&ws[(size_t)img * WS_PER_IMG + 32 + tid], s);
    }
}

// ---------------------------------------------------------------------------
// Finalize: one wave, lane k <-> instance k. O(K^2 * D) epilogue.
// ---------------------------------------------------------------------------
__global__ __launch_bounds__(32) void finalize_kernel(
        const float* __restrict__ ws, float* __restrict__ out)
{
    __shared__ float s_mu[K_IDS][DIMS + 1];  // +1 pad vs bank conflicts
    __shared__ int   s_pres[K_IDS];

    const int k = threadIdx.x;  // 0..31
    float batch_acc = 0.0f;

    for (int img = 0; img < NIMG; ++img) {
        const float* w  = ws + (size_t)img * WS_PER_IMG;
        const float cnt = w[k];
        const float ssq = w[32 + k];
        const bool  present = (cnt > 0.0f) && (k != 0);
        const float c1 = (cnt > 0.0f) ? cnt : 1.0f;

        float mu[DIMS];
        float musq = 0.0f;
#pragma unroll
        for (int d = 0; d < DIMS; ++d) {
            mu[d] = w[64 + k * DIMS + d] / c1;
            musq += mu[d] * mu[d];
        }

        __syncthreads();  // done with previous image's s_mu
#pragma unroll
        for (int d = 0; d < DIMS; ++d) s_mu[k][d] = mu[d];
        s_pres[k] = present ? 1 : 0;
        __syncthreads();

        // variance term: sum||f-mu||^2 / c = ssq/c - ||mu||^2
        float var_t = present ? (ssq / c1 - musq) : 0.0f;
        if (var_t < 0.0f) var_t = 0.0f;  // guard fp rounding
        float reg_t    = present ? sqrtf(musq) : 0.0f;
        float nclust_t = present ? 1.0f : 0.0f;

        float pair_t = 0.0f;
        for (int j = k + 1; j < K_IDS; ++j) {
            if (present && s_pres[j]) {
                float d2 = 0.0f;
#pragma unroll
                for (int d = 0; d < DIMS; ++d) {
                    float df = mu[d] - s_mu[j][d];
                    d2 += df * df;
                }
                float dist = sqrtf(d2);
                float hg = 2.0f * DD_C - dist;
                if (hg > 0.0f) pair_t += hg * hg;
            }
        }

        // full-wave reduction (wave32)
#pragma unroll
        for (int o = 16; o >= 1; o >>= 1) {
            var_t    += __shfl_xor(var_t,    o, 32);
            reg_t    += __shfl_xor(reg_t,    o, 32);
            pair_t   += __shfl_xor(pair_t,   o, 32);
            nclust_t += __shfl_xor(nclust_t, o, 32);
        }

        const float nclust = nclust_t;
        const float denom = fmaxf(nclust - 1.0f, 1.0f);
        const float dist_loss = (nclust > 1.0f) ? (pair_t / denom) : 0.0f;
        const float total = (var_t + dist_loss + GAMMA_C * reg_t)
                            / fmaxf(nclust, 1.0f);
        batch_acc += (nclust > 0.0f) ? total : 0.0f;
    }

    if (k == 0) out[0] = batch_acc / (float)(NIMG + 1);
}

// ---------------------------------------------------------------------------

extern "C" void kernel_launch(void* const* d_in, const int* in_sizes, int n_in,
                              void* d_out, int out_size, void* d_ws, size_t ws_size,
                              hipStream_t stream) {
    const float* F   = (const float*)d_in[0];  // [8,32,512,512] f32
    const int*   Lab = (const int*)  d_in[1];  // [8,512,512]    i32
    float*       ws  = (float*)d_ws;
    float*       out = (float*)d_out;

    const int nz = NIMG * WS_PER_IMG;
    zero_ws_kernel<<<(nz + 255) / 256, 256, 0, stream>>>(ws, nz);

    dim3 grid(BLOCKS_PER_IMG, NIMG);
    segsum_wmma_kernel<<<grid, 256, 0, stream>>>(F, Lab, ws);

    finalize_kernel<<<1, 32, 0, stream>>>(ws, out);
}